// SegmentMotionConsistencyLoss_48318382080471
// MI455X (gfx1250) — compile-verified
//
#include <hip/hip_runtime.h>
#include <hip/hip_bf16.h>
#include <math.h>

// Problem constants (from reference setup_inputs)
#define BATCH 8
#define NSEG  24
#define HW    (512 * 512)
#define TILE  64          // pixels per staged LDS tile
#define SEGP  32          // padded segment rows in LDS (24 real + 8 zero)

typedef __attribute__((ext_vector_type(2))) float v2f;
typedef __attribute__((ext_vector_type(8))) float v8f;

// ---------------------------------------------------------------------------
// zero scratch accumulators: acc[B*S*3] then num[S]
// ---------------------------------------------------------------------------
__global__ void SegLoss_zero_kernel(float* __restrict__ ws, int n) {
    int i = blockIdx.x * 256 + threadIdx.x;
    if (i < n) ws[i] = 0.0f;
}

// ---------------------------------------------------------------------------
// Pass 1: per (b,s) compute [sum(mask), sum(mask*fx), sum(mask*fy)] using
// V_WMMA_F32_16X16X4_F32.  A = mask tile [16 seg x 4 px], B = [4 px x 16]
// with columns {1, fx, fy, 0...}.  D accumulates [16 seg x 16], cols 0..2 used.
// ---------------------------------------------------------------------------
__global__ __launch_bounds__(256) void SegLoss_pass1_wmma_kernel(
        const float* __restrict__ flow,     // [B,2,H,W]
        const float* __restrict__ masks,    // [B,S,H,W]
        float* __restrict__ acc)            // [B,S,3]
{
    __shared__ float lm[SEGP * TILE];   // mask tile, rows 24..31 zeroed
    __shared__ float lf[2 * TILE];      // fx then fy

    const int b    = blockIdx.y;
    const int tid  = threadIdx.x;
    const int wave = tid >> 5;
    const int lane = tid & 31;
    const int segTile = wave & 1;        // 0: segs 0-15, 1: segs 16-31 (padded)
    const int pslice  = wave >> 1;       // 0..3 -> 16-pixel slice of the tile
    const int m       = lane & 15;       // A row (segment) == B column index
    const int khalf   = (lane >> 4) << 1; // lanes 0-15 hold K=0/1, 16-31 hold K=2/3

    const float* maskB = masks + (size_t)b * NSEG * HW;
    const float* flowB = flow  + (size_t)b * 2 * HW;

    v8f c = {};  // fp32 accumulator (16x16 D matrix, 8 VGPRs)

    const int nchunk = HW / TILE;  // 4096, uniform across the grid
    for (int chunk = blockIdx.x; chunk < nchunk; chunk += gridDim.x) {
        const int pbase = chunk * TILE;

        // ---- coalesced cooperative load into LDS ----
        for (int t = tid; t < SEGP * TILE; t += 256) {
            int s = t / TILE, p = t - s * TILE;
            lm[t] = (s < NSEG) ? maskB[(size_t)s * HW + pbase + p] : 0.0f;
        }
        for (int t = tid; t < 2 * TILE; t += 256) {
            int ch = t / TILE, p = t - ch * TILE;
            lf[t] = flowB[(size_t)ch * HW + pbase + p];
        }
        __syncthreads();

        // ---- 4 WMMA steps covering this wave's 16 pixels ----
        const int segBase = segTile * 16;
#pragma unroll
        for (int kk = 0; kk < 4; ++kk) {
            const int k0 = pslice * 16 + kk * 4 + khalf;  // pixel within tile
            v2f a;  // A: mask[segBase+m][k0], [k0+1]
            a.x = lm[(segBase + m) * TILE + k0];
            a.y = lm[(segBase + m) * TILE + k0 + 1];
            // B: row k, column n=m: n==0 -> 1, n==1 -> fx[k], n==2 -> fy[k]
            float fx0 = lf[k0],     fy0 = lf[TILE + k0];
            float fx1 = lf[k0 + 1], fy1 = lf[TILE + k0 + 1];
            v2f bb;
            bb.x = (m == 0) ? 1.0f : (m == 1) ? fx0 : (m == 2) ? fy0 : 0.0f;
            bb.y = (m == 0) ? 1.0f : (m == 1) ? fx1 : (m == 2) ? fy1 : 0.0f;
            c = __builtin_amdgcn_wmma_f32_16x16x4_f32(
                    false, a, false, bb, (short)0, c, false, false);
        }
        __syncthreads();
    }

    // ---- writeback: D element [r + (lane<16?0:8)][n=lane&15] in c[r] ----
    const int n = lane & 15;
    if (n < 3) {
        const int rbase = segTile * 16 + ((lane >> 4) << 3);
#pragma unroll
        for (int r = 0; r < 8; ++r) {
            int seg = rbase + r;
            if (seg < NSEG)
                atomicAdd(&acc[(b * NSEG + seg) * 3 + n], c[r]);
        }
    }
}

// ---------------------------------------------------------------------------
// Pass 2: num[s] += sum_{b,h,w} (|flow - mean_s|^2 + eps^2)^0.45 * mask
// ---------------------------------------------------------------------------
__global__ __launch_bounds__(256) void SegLoss_pass2_kernel(
        const float* __restrict__ flow,
        const float* __restrict__ masks,
        const float* __restrict__ acc,   // [B,S,3] from pass 1
        float* __restrict__ num)         // [S]
{
    __shared__ float smean[NSEG][2];
    __shared__ float lnum[NSEG];

    const int b   = blockIdx.y;
    const int tid = threadIdx.x;
    const int lane = tid & 31;

    if (tid < NSEG) {
        float ms = fmaxf(acc[(b * NSEG + tid) * 3 + 0], 1e-6f);
        smean[tid][0] = acc[(b * NSEG + tid) * 3 + 1] / ms;
        smean[tid][1] = acc[(b * NSEG + tid) * 3 + 2] / ms;
        lnum[tid] = 0.0f;
    }
    __syncthreads();

    // hoist means into registers
    float mx[NSEG], my[NSEG], a[NSEG];
#pragma unroll
    for (int s = 0; s < NSEG; ++s) {
        mx[s] = smean[s][0];
        my[s] = smean[s][1];
        a[s]  = 0.0f;
    }

    const float* maskB = masks + (size_t)b * NSEG * HW;
    const float* flowB = flow  + (size_t)b * 2 * HW;

    for (int p = blockIdx.x * 256 + tid; p < HW; p += gridDim.x * 256) {
        float fx = flowB[p];
        float fy = flowB[HW + p];
#pragma unroll
        for (int s = 0; s < NSEG; ++s) {
            float dx = fx - mx[s];
            float dy = fy - my[s];
            float mag2 = fmaf(dx, dx, dy * dy);
            // charbonnier: (mag2 + (1e-3)^2)^0.45, arg > 0 always
            float ch = exp2f(0.45f * log2f(mag2 + 1e-6f));
            a[s] = fmaf(ch, maskB[(size_t)s * HW + p], a[s]);  // coalesced
        }
    }

    // wave32 reduction, then LDS, then one atomic per (block, s)
#pragma unroll
    for (int s = 0; s < NSEG; ++s) {
#pragma unroll
        for (int off = 16; off > 0; off >>= 1)
            a[s] += __shfl_down(a[s], off, 32);
        if (lane == 0) atomicAdd(&lnum[s], a[s]);
    }
    __syncthreads();
    if (tid < NSEG) atomicAdd(&num[tid], lnum[tid]);
}

// ---------------------------------------------------------------------------
// Finalize: match reference clip/guard semantics exactly
// ---------------------------------------------------------------------------
__global__ void SegLoss_final_kernel(const float* __restrict__ acc,
                                     const float* __restrict__ num,
                                     float* __restrict__ out)
{
    if (threadIdx.x == 0) {
        float total_loss = 0.0f, total_w = 0.0f;
        for (int s = 0; s < NSEG; ++s) {
            float segw = 0.0f;
            for (int b = 0; b < BATCH; ++b)
                segw += fmaxf(acc[(b * NSEG + s) * 3 + 0], 1e-6f);  // clipped mask_sum
            float variance = num[s] / fmaxf(segw, 1e-6f);
            total_loss += variance * segw;
            total_w    += segw;
        }
        out[0] = (total_w > 1e-6f) ? 0.05f * (total_loss / total_w) : 0.0f;
    }
}

// ---------------------------------------------------------------------------
extern "C" void kernel_launch(void* const* d_in, const int* in_sizes, int n_in,
                              void* d_out, int out_size, void* d_ws, size_t ws_size,
                              hipStream_t stream) {
    const float* flow  = (const float*)d_in[0];   // [8,2,512,512]
    const float* masks = (const float*)d_in[1];   // [8,24,512,512]
    float* acc = (float*)d_ws;                    // B*S*3 floats
    float* num = acc + BATCH * NSEG * 3;          // S floats
    float* out = (float*)d_out;

    const int nz = BATCH * NSEG * 3 + NSEG;       // 600
    SegLoss_zero_kernel<<<dim3((nz + 255) / 256), 256, 0, stream>>>(acc, nz);
    SegLoss_pass1_wmma_kernel<<<dim3(512, BATCH), 256, 0, stream>>>(flow, masks, acc);
    SegLoss_pass2_kernel<<<dim3(256, BATCH), 256, 0, stream>>>(flow, masks, acc, num);
    SegLoss_final_kernel<<<1, 32, 0, stream>>>(acc, num, out);
}